// CDHMM_79173427134659
// MI455X (gfx1250) — compile-verified
//
#include <hip/hip_runtime.h>
#include <math.h>
#include <stdint.h>

#define T_LEN  4096
#define D_DIM  64
#define K_Q    16
#define N_Z    16
#define M_MIX  8
#define NSTATE 256   // K_Q * N_Z

typedef __attribute__((ext_vector_type(2))) float v2f;
typedef __attribute__((ext_vector_type(8))) float v8f;

__device__ __forceinline__ float digamma_dev(float x){
  if (!(x > 0.0f)) return __builtin_nanf("");
  float r = 0.0f;
  while (x < 6.0f){ r -= 1.0f/x; x += 1.0f; }   // terminates in <= 7 iters for x>0
  float inv  = 1.0f/x;
  float inv2 = inv*inv;
  return r + logf(x) - 0.5f*inv
       - inv2*(1.0f/12.0f - inv2*(1.0f/120.0f - inv2*(1.0f/252.0f)));
}

// ---------------------------------------------------------------------------
// Kernel 1: per-(n,m) Cholesky of t_S (64x64 SPD), L^{-1}, and half_log_det.
// One block per (n,m), 64 threads, matrix staged in LDS.
// ---------------------------------------------------------------------------
__global__ __launch_bounds__(64) void cdhmm_chol(const float* __restrict__ tS,
                                                 float* __restrict__ Linv,
                                                 float* __restrict__ hld){
  __shared__ float A[D_DIM*D_DIM];                 // 16 KB
  const int nm  = blockIdx.x;
  const int tid = threadIdx.x;
  for (int r = 0; r < D_DIM; ++r)
    A[r*D_DIM + tid] = tS[(size_t)nm*D_DIM*D_DIM + (size_t)r*D_DIM + tid];
  __syncthreads();

  for (int k = 0; k < D_DIM; ++k){
    if (tid == 0) A[k*D_DIM + k] = sqrtf(A[k*D_DIM + k]);
    __syncthreads();
    if (tid > k) A[tid*D_DIM + k] /= A[k*D_DIM + k];
    __syncthreads();
    if (tid > k){
      float lik = A[tid*D_DIM + k];
      for (int j = k + 1; j <= tid; ++j) A[tid*D_DIM + j] -= lik * A[j*D_DIM + k];
    }
    __syncthreads();
  }
  if (tid == 0){
    float s = 0.0f;
    for (int d = 0; d < D_DIM; ++d) s += logf(A[d*D_DIM + d]);
    hld[nm] = s;
  }
  // forward substitution: thread tid computes column tid of L^{-1}
  float y[D_DIM];
  for (int r = 0; r < D_DIM; ++r){
    if (r < tid){ y[r] = 0.0f; continue; }
    float acc = (r == tid) ? 1.0f : 0.0f;
    for (int j = tid; j < r; ++j) acc -= A[r*D_DIM + j] * y[j];
    y[r] = acc / A[r*D_DIM + r];
  }
  for (int r = 0;   r < tid;   ++r) Linv[(size_t)nm*4096 + r*D_DIM + tid] = 0.0f;
  for (int r = tid; r < D_DIM; ++r) Linv[(size_t)nm*4096 + r*D_DIM + tid] = y[r];
}

// ---------------------------------------------------------------------------
// Kernel 2: digamma-based constants and the exp(LT) transition matrix (+T).
// One block, 256 threads.
// ---------------------------------------------------------------------------
__global__ __launch_bounds__(256) void cdhmm_prep(const float* __restrict__ t_l,
     const float* __restrict__ t_eta, const float* __restrict__ t_w,
     const float* __restrict__ t_pi, const float* __restrict__ t_A,
     const float* __restrict__ t_hat_A, const float* __restrict__ t_hat_pi,
     const float* __restrict__ hld,
     float* __restrict__ cterm, float* __restrict__ elogw,
     float* __restrict__ Mg, float* __restrict__ MTg, float* __restrict__ misc){
  __shared__ float logA[K_Q*N_Z*N_Z];   // 16 KB
  __shared__ float hAe[K_Q*K_Q];        // 1 KB
  const int tid = threadIdx.x;
  {
    const int k = tid >> 4, n = tid & 15;
    const float* row = t_A + (size_t)(k*N_Z + n)*N_Z;
    float s = 0.0f;
    for (int q = 0; q < 16; ++q) s += row[q];
    float ds = digamma_dev(s);
    for (int q = 0; q < 16; ++q) logA[(k*16 + n)*16 + q] = digamma_dev(row[q]) - ds;
  }
  {
    const int k = tid >> 4, k2 = tid & 15;
    const float* row = t_hat_A + k*16;
    float s = 0.0f;
    for (int q = 0; q < 16; ++q) s += row[q];
    hAe[tid] = digamma_dev(row[k2]) - digamma_dev(s);
  }
  if (tid < 16){
    float s = 0.0f;  for (int q = 0; q < 16; ++q) s  += t_pi[q];
    misc[tid] = digamma_dev(t_pi[tid]) - digamma_dev(s);
    float s2 = 0.0f; for (int q = 0; q < 16; ++q) s2 += t_hat_pi[q];
    misc[16 + tid] = digamma_dev(t_hat_pi[tid]) - digamma_dev(s2);
  }
  if (tid < 128){
    const int n = tid >> 3;
    const float* row = t_w + n*M_MIX;
    float s = 0.0f;
    for (int q = 0; q < M_MIX; ++q) s += row[q];
    elogw[tid] = digamma_dev(t_w[tid]) - digamma_dev(s);
    float eta = t_eta[tid];
    float er = 0.0f;
    for (int d = 1; d <= D_DIM; ++d) er += digamma_dev(0.5f*(eta + 1.0f - (float)d));
    if (!isfinite(er)) er = 0.0f;
    float term1 = er + 64.0f*0.69314718055994531f + 2.0f*hld[tid];
    cterm[tid] = 0.5f*term1 - 32.0f*logf(6.28318530717958648f) - 32.0f/t_l[tid];
  }
  __syncthreads();
  {
    // M[(k,n)][(k',n')] = exp(hat_A_e[k,k'] + log_A_e[k', n, n'])
    const int k = tid >> 4, n = tid & 15;
    for (int k2 = 0; k2 < 16; ++k2){
      float h = hAe[k*16 + k2];
      for (int n2 = 0; n2 < 16; ++n2){
        float val = __expf(h + logA[(k2*16 + n)*16 + n2]);
        const int jj = k2*16 + n2;
        Mg [(size_t)tid*NSTATE + jj] = val;
        MTg[(size_t)jj*NSTATE + tid] = val;
      }
    }
  }
}

// ---------------------------------------------------------------------------
// Kernel 3: emission loglik lp[T,N] via WMMA f32 16x16x4.
// Grid: T/16 blocks x 256 threads (8 waves). Wave handles 2 n's x 8 m's.
// C tile = Linv(16x4 chunk) x diff^T(4x16 t-chunk), K-accumulated over D=64.
// X tile is staged into LDS with the CDNA5 async-to-LDS engine (ASYNCcnt).
// ---------------------------------------------------------------------------
__global__ __launch_bounds__(256) void cdhmm_emis(const float* __restrict__ X,
     const float* __restrict__ t_m, const float* __restrict__ t_eta,
     const float* __restrict__ Linv, const float* __restrict__ cterm,
     const float* __restrict__ elogw, float* __restrict__ lp){
  __shared__ float Xs[16*65];          // padded to dodge bank conflicts
  const int t0  = blockIdx.x * 16;
  const int tid = threadIdx.x;
  // Async global->LDS staging: each thread copies 4 B32 slots directly into
  // LDS without touching VGPRs (tracked by ASYNCcnt, not LOADcnt).
  for (int idx = tid; idx < 16*64; idx += 256){
    int r = idx >> 6, c = idx & 63;
    uint32_t laddr = (uint32_t)(uintptr_t)(&Xs[r*65 + c]);
    uint64_t gaddr = (uint64_t)(uintptr_t)(&X[(size_t)(t0 + r)*D_DIM + c]);
    asm volatile("global_load_async_to_lds_b32 %0, %1, off"
                 :: "v"(laddr), "v"(gaddr) : "memory");
  }
  asm volatile("s_wait_asynccnt 0x0" ::: "memory");
  __syncthreads();
  const int wave  = tid >> 5;
  const int lane  = tid & 31;
  const int l16   = lane & 15;
  const int lhalf = lane >> 4;
  for (int nh = 0; nh < 2; ++nh){
    const int n = wave*2 + nh;
    float runmax = -INFINITY, runsum = 0.0f;
    for (int m = 0; m < M_MIX; ++m){
      const int nm = n*M_MIX + m;
      const float* Lv = Linv + (size_t)nm*4096;
      const float* mu = t_m  + (size_t)nm*64;
      v8f acc0 = {}, acc1 = {}, acc2 = {}, acc3 = {};
      for (int kk = 0; kk < 16; ++kk){
        const int jj = kk*4 + lhalf*2;           // K columns owned by this half-wave
        v2f b;                                   // B(4x16): K striped over {vgpr,half}, N over lanes
        b.x = Xs[l16*65 + jj]     - mu[jj];
        b.y = Xs[l16*65 + jj + 1] - mu[jj + 1];
        v2f a;                                   // A(16x4): M over lanes, K over {vgpr,half}
        a.x = Lv[( 0 + l16)*64 + jj];  a.y = Lv[( 0 + l16)*64 + jj + 1];
        acc0 = __builtin_amdgcn_wmma_f32_16x16x4_f32(false, a, false, b, (short)0, acc0, false, false);
        a.x = Lv[(16 + l16)*64 + jj];  a.y = Lv[(16 + l16)*64 + jj + 1];
        acc1 = __builtin_amdgcn_wmma_f32_16x16x4_f32(false, a, false, b, (short)0, acc1, false, false);
        a.x = Lv[(32 + l16)*64 + jj];  a.y = Lv[(32 + l16)*64 + jj + 1];
        acc2 = __builtin_amdgcn_wmma_f32_16x16x4_f32(false, a, false, b, (short)0, acc2, false, false);
        a.x = Lv[(48 + l16)*64 + jj];  a.y = Lv[(48 + l16)*64 + jj + 1];
        acc3 = __builtin_amdgcn_wmma_f32_16x16x4_f32(false, a, false, b, (short)0, acc3, false, false);
      }
      // maha[t]: sum of squares over all 64 rows; lane pair (l, l^16) covers all rows of column t
      float ss = 0.0f;
      #pragma unroll
      for (int r = 0; r < 8; ++r)
        ss += acc0[r]*acc0[r] + acc1[r]*acc1[r] + acc2[r]*acc2[r] + acc3[r]*acc3[r];
      ss += __shfl_xor(ss, 16, 32);
      float llv = cterm[nm] - 0.5f * t_eta[nm] * ss;
      if (!isfinite(llv)) llv = 1.0f;
      llv += elogw[nm];
      if (llv > runmax){ runsum = runsum * __expf(runmax - llv) + 1.0f; runmax = llv; }
      else             { runsum += __expf(llv - runmax); }
    }
    if (lane < 16) lp[(size_t)(t0 + lane)*N_Z + n] = runmax + logf(runsum);
  }
}

// ---------------------------------------------------------------------------
// Kernel 4: scaled forward pass (sequential over T). 1 block x 256 threads.
// Writes normalized alpha_hat[T,256] and ll (=sum of log normalizers).
// ---------------------------------------------------------------------------
__global__ __launch_bounds__(256) void cdhmm_fwd(const float* __restrict__ lp,
     const float* __restrict__ Mg, const float* __restrict__ misc,
     float* __restrict__ alpha_ws, float* __restrict__ ll_out){
  __shared__ float al[NSTATE];
  __shared__ float red[NSTATE];
  const int j = threadIdx.x;
  const int n = j & 15, k = j >> 4;

  float mx = -INFINITY;
  for (int q = 0; q < 16; ++q) mx = fmaxf(mx, lp[q]);
  float u = __expf(misc[16 + k] + misc[n] + lp[n] - mx);
  red[j] = u; __syncthreads();
  for (int s = 128; s > 0; s >>= 1){ if (j < s) red[j] += red[j + s]; __syncthreads(); }
  float c = red[0]; __syncthreads();
  float logZ = logf(c) + mx;
  al[j] = u / c; alpha_ws[j] = al[j];
  __syncthreads();

  for (int t = 1; t < T_LEN; ++t){
    const float* lpt = lp + t*16;
    __builtin_prefetch(lp + (t + 1)*16, 0, 0);
    float mx2 = -INFINITY;
    for (int q = 0; q < 16; ++q) mx2 = fmaxf(mx2, lpt[q]);
    const float* col = Mg + j;                    // column j of M, coalesced across lanes
    float s = 0.0f;
    for (int i = 0; i < NSTATE; i += 4){
      s += al[i+0]*col[(i+0)*NSTATE] + al[i+1]*col[(i+1)*NSTATE]
         + al[i+2]*col[(i+2)*NSTATE] + al[i+3]*col[(i+3)*NSTATE];
    }
    float u2 = s * __expf(lpt[n] - mx2);
    red[j] = u2; __syncthreads();
    for (int sr = 128; sr > 0; sr >>= 1){ if (j < sr) red[j] += red[j + sr]; __syncthreads(); }
    float c2 = red[0]; __syncthreads();
    logZ += logf(c2) + mx2;
    al[j] = u2 / c2;
    alpha_ws[(size_t)t*NSTATE + j] = al[j];
    __syncthreads();
  }
  if (j == 0) ll_out[0] = logZ;
}

// ---------------------------------------------------------------------------
// Kernel 5: backward pass fused with gamma and xi accumulation.
// xi_t[i,j] = al_t[i] * M[i,j] * p_{t+1}[j] * bh_{t+1}[j] / S_t, sums to 1.
// Thread j owns target state (k',n'): trans_z partials are conflict-free
// per-thread registers; trans_q reduced deterministically at the end.
// ---------------------------------------------------------------------------
__global__ __launch_bounds__(256) void cdhmm_bwd(const float* __restrict__ lp,
     const float* __restrict__ Mg, const float* __restrict__ MTg,
     const float* __restrict__ alpha_ws,
     float* __restrict__ gamma_out, float* __restrict__ tz_out,
     float* __restrict__ tq_out){
  __shared__ float bh[NSTATE], v[NSTATE], al[NSTATE], red[NSTATE];
  const int j  = threadIdx.x;
  const int np = j & 15, kp = j >> 4;
  float tzacc[16], tqacc[16];
  #pragma unroll
  for (int q = 0; q < 16; ++q){ tzacc[q] = 0.0f; tqacc[q] = 0.0f; }

  bh[j] = 1.0f/256.0f;                                   // beta_hat(T-1), any uniform scale
  al[j] = alpha_ws[(size_t)(T_LEN - 1)*NSTATE + j];
  gamma_out[(size_t)(T_LEN - 1)*NSTATE + j] = al[j];     // uniform beta -> gamma = alpha_hat
  __syncthreads();

  for (int t = T_LEN - 2; t >= 0; --t){
    const float* lpn = lp + (t + 1)*16;
    __builtin_prefetch(lp + t*16, 0, 0);
    float mx = -INFINITY;
    for (int q = 0; q < 16; ++q) mx = fmaxf(mx, lpn[q]);
    v[j]  = __expf(lpn[np] - mx) * bh[j];
    al[j] = alpha_ws[(size_t)t*NSTATE + j];
    __syncthreads();

    // beta row-dot rd[i] = sum_j M[i,j] v[j], via MT for coalescing
    const float* mtcol = MTg + j;
    float rd = 0.0f;
    for (int q = 0; q < NSTATE; q += 4){
      rd += mtcol[(q+0)*NSTATE]*v[q+0] + mtcol[(q+1)*NSTATE]*v[q+1]
          + mtcol[(q+2)*NSTATE]*v[q+2] + mtcol[(q+3)*NSTATE]*v[q+3];
    }
    red[j] = al[j]*rd; __syncthreads();
    for (int s = 128; s > 0; s >>= 1){ if (j < s) red[j] += red[j + s]; __syncthreads(); }
    float S = red[0]; __syncthreads();
    float invS = 1.0f / S;
    gamma_out[(size_t)t*NSTATE + j] = al[j]*rd*invS;

    red[j] = rd; __syncthreads();
    for (int s = 128; s > 0; s >>= 1){ if (j < s) red[j] += red[j + s]; __syncthreads(); }
    float cb = red[0]; __syncthreads();

    // xi accumulation for this thread's target state (kp, np)
    const float* mcol = Mg + j;
    float w = v[j] * invS;
    #pragma unroll
    for (int nn2 = 0; nn2 < 16; ++nn2){
      float accn = 0.0f;
      #pragma unroll
      for (int kq = 0; kq < 16; ++kq){
        float pr = al[kq*16 + nn2] * mcol[(kq*16 + nn2)*NSTATE];
        accn     += pr;
        tqacc[kq] += pr * w;
      }
      tzacc[nn2] += accn * w;          // trans_z[kp, nn2, np] partial
    }
    bh[j] = rd * (1.0f / cb);
    __syncthreads();
  }

  #pragma unroll
  for (int nn2 = 0; nn2 < 16; ++nn2)
    tz_out[kp*256 + nn2*16 + np] = tzacc[nn2];

  #pragma unroll
  for (int kq = 0; kq < 16; ++kq){
    red[j] = tqacc[kq]; __syncthreads();
    if (np == 0){
      float s = 0.0f;
      #pragma unroll
      for (int q = 0; q < 16; ++q) s += red[kp*16 + q];
      tq_out[kq*16 + kp] = s;
    }
    __syncthreads();
  }
}

// ---------------------------------------------------------------------------
extern "C" void kernel_launch(void* const* d_in, const int* in_sizes, int n_in,
                              void* d_out, int out_size, void* d_ws, size_t ws_size,
                              hipStream_t stream){
  (void)in_sizes; (void)n_in; (void)out_size; (void)ws_size;
  const float* X       = (const float*)d_in[0];
  const float* t_l     = (const float*)d_in[1];
  const float* t_m     = (const float*)d_in[2];
  const float* t_eta   = (const float*)d_in[3];
  const float* t_S     = (const float*)d_in[4];
  const float* t_w     = (const float*)d_in[5];
  const float* t_pi    = (const float*)d_in[6];
  const float* t_A     = (const float*)d_in[7];
  const float* t_hatA  = (const float*)d_in[8];
  const float* t_hatpi = (const float*)d_in[9];

  float* ws    = (float*)d_ws;
  float* Linv  = ws;                       // 524288 floats (2 MB)
  float* Mg    = Linv  + 524288;           // 65536
  float* MTg   = Mg    + 65536;            // 65536
  float* lp    = MTg   + 65536;            // 65536
  float* alpha = lp    + 65536;            // 1048576 (4 MB)
  float* hld   = alpha + 1048576;          // 128
  float* cterm = hld   + 128;              // 128
  float* elogw = cterm + 128;              // 128
  float* misc  = elogw + 128;              // 64
  // total ~7.1 MB of workspace

  float* out       = (float*)d_out;
  float* ll_out    = out;                                   // [1]
  float* gamma_out = out + 1;                               // [T,K,N]
  float* tz_out    = gamma_out + (size_t)T_LEN*NSTATE;      // [K,N,N]
  float* tq_out    = tz_out + K_Q*N_Z*N_Z;                  // [K,K]

  cdhmm_chol<<<dim3(N_Z*M_MIX), dim3(64),  0, stream>>>(t_S, Linv, hld);
  cdhmm_prep<<<dim3(1),          dim3(256), 0, stream>>>(t_l, t_eta, t_w, t_pi, t_A,
                                                         t_hatA, t_hatpi, hld,
                                                         cterm, elogw, Mg, MTg, misc);
  cdhmm_emis<<<dim3(T_LEN/16),   dim3(256), 0, stream>>>(X, t_m, t_eta, Linv, cterm, elogw, lp);
  cdhmm_fwd <<<dim3(1),          dim3(256), 0, stream>>>(lp, Mg, misc, alpha, ll_out);
  cdhmm_bwd <<<dim3(1),          dim3(256), 0, stream>>>(lp, Mg, MTg, alpha,
                                                         gamma_out, tz_out, tq_out);
}